// DoG_Seasonal_57002805952721
// MI455X (gfx1250) — compile-verified
//
#include <hip/hip_runtime.h>

typedef __attribute__((ext_vector_type(2))) float v2f;
typedef __attribute__((ext_vector_type(8))) float v8f;

#define NB 32
#define NL 4096
#define NC 321
#define R2 384
#define KW 769                 // combined DoG kernel taps
#define PAD_LO 63
#define SWC_LEN (KW + 2 * PAD_LO)   // zero-padded so A lookups need no bounds check
#define NKK 208                // (64 + 2*384)/4 K-chunks per 64-row super-tile
#define CTILES 21              // ceil(321/16)
#define LTILES (NL / 64)       // 64
#define NTILES (NB * LTILES * CTILES)  // 43008 waves
#define WAVES 8                // 256-thread blocks

// ---------------- Build combined DoG weights (deterministic) ----------------
__global__ __launch_bounds__(512) void build_wc(float* __restrict__ wc) {
    __shared__ float s1[512], s2[512];
    __shared__ float inv1, inv2;
    const int t = threadIdx.x;
    float p1 = 0.f, p2 = 0.f;
    for (int d = t; d < KW; d += 512) {
        float u = (float)(d - R2);
        p2 += expf(-0.5f * (u / 96.0f) * (u / 96.0f));
        if (u >= -17.0f && u <= 17.0f)
            p1 += expf(-0.5f * (u / 4.2f) * (u / 4.2f));
    }
    s1[t] = p1; s2[t] = p2;
    __syncthreads();
    if (t == 0) {  // fixed-order serial reduction => bit-deterministic
        float a = 0.f, c = 0.f;
        for (int i = 0; i < 512; ++i) { a += s1[i]; c += s2[i]; }
        inv1 = 1.0f / a; inv2 = 1.0f / c;
    }
    __syncthreads();
    for (int d = t; d < KW; d += 512) {
        float u = (float)(d - R2);
        float g2 = expf(-0.5f * (u / 96.0f) * (u / 96.0f)) * inv2;
        float g1 = (u >= -17.0f && u <= 17.0f)
                       ? expf(-0.5f * (u / 4.2f) * (u / 4.2f)) * inv1 : 0.f;
        wc[d] = g1 - g2;
    }
}

// A fragments for the 4 stacked L-tiles (Toeplitz: shift 16/tile) + 4 WMMAs.
// Kept as a macro so `swc` stays a known-LDS access (ds_load, not flat).
#define CHUNK_MMA(A0IDX, BV)                                                        \
    {                                                                               \
        const int _a0 = (A0IDX);                                                    \
        v2f _A0, _A1, _A2, _A3;                                                     \
        _A0.x = swc[_a0];      _A0.y = swc[_a0 + 1];                                \
        _A1.x = swc[_a0 - 16]; _A1.y = swc[_a0 - 15];                               \
        _A2.x = swc[_a0 - 32]; _A2.y = swc[_a0 - 31];                               \
        _A3.x = swc[_a0 - 48]; _A3.y = swc[_a0 - 47];                               \
        acc0 = __builtin_amdgcn_wmma_f32_16x16x4_f32(false, _A0, false, (BV),       \
                                                     (short)0, acc0, false, false); \
        acc1 = __builtin_amdgcn_wmma_f32_16x16x4_f32(false, _A1, false, (BV),       \
                                                     (short)0, acc1, false, false); \
        acc2 = __builtin_amdgcn_wmma_f32_16x16x4_f32(false, _A2, false, (BV),       \
                                                     (short)0, acc2, false, false); \
        acc3 = __builtin_amdgcn_wmma_f32_16x16x4_f32(false, _A3, false, (BV),       \
                                                     (short)0, acc3, false, false); \
    }

// ---------------- Main DoG via banded-Toeplitz f32 WMMA ----------------
__global__ __launch_bounds__(256) void dog_wmma(const float* __restrict__ x,
                                                const float* __restrict__ wc,
                                                float* __restrict__ out) {
    __shared__ float swc[SWC_LEN];
    for (int i = threadIdx.x; i < SWC_LEN; i += 256) {
        int d = i - PAD_LO;
        swc[i] = (d >= 0 && d < KW) ? wc[d] : 0.0f;
    }
    __syncthreads();

    const int lane = threadIdx.x & 31;
    const int wave = threadIdx.x >> 5;
    int tile = blockIdx.x * WAVES + wave;          // exact grid: no guard needed
    const int ct = tile % CTILES;  tile /= CTILES;
    const int lt = tile % LTILES;
    const int b  = tile / LTILES;

    const int n  = lane & 15;      // channel offset == A row m (both lane&15)
    const int kh = lane >> 4;      // which K-half this lane holds
    const int c0 = ct * 16;
    int cidx = c0 + n;
    if (cidx >= NC) cidx = NC - 1;                 // clamp: keep EXEC full in loop
    const int l0 = lt * 64;

    const float* xb = x + (size_t)b * ((size_t)NL * NC);
    const int abase = PAD_LO + kh * 2 - n;         // A lookup base (kk=0, tile 0)
    const int pbase = l0 - R2 + kh * 2;            // input position base

    v8f acc0 = {0,0,0,0,0,0,0,0};
    v8f acc1 = {0,0,0,0,0,0,0,0};
    v8f acc2 = {0,0,0,0,0,0,0,0};
    v8f acc3 = {0,0,0,0,0,0,0,0};

    // Window spans [l0-384, l0+447]; interior tiles never reflect.
    const bool interior = (l0 >= R2) && (l0 + 63 + R2 < NL);  // wave-uniform

    if (interior) {
        // Fast path: monotone addresses, pointer bump of 4 rows per chunk,
        // second K-row at a constant byte offset => no per-chunk 64-bit MADs.
        const float* px = xb + (size_t)pbase * NC + cidx;
#pragma unroll 4
        for (int kk = 0; kk < NKK; ++kk) {
            v2f bv;
            bv.x = px[0];
            bv.y = px[NC];
            px += 4 * (size_t)NC;
            CHUNK_MMA(abase + kk * 4, bv);
        }
    } else {
        // Boundary path: reflect-padded gather.
#pragma unroll 4
        for (int kk = 0; kk < NKK; ++kk) {
            const int p0 = pbase + kk * 4;
            const int p1 = p0 + 1;
            int rp0 = p0 < 0 ? -p0 : (p0 >= NL ? 2 * NL - 2 - p0 : p0);
            int rp1 = p1 < 0 ? -p1 : (p1 >= NL ? 2 * NL - 2 - p1 : p1);
            v2f bv;
            bv.x = xb[(size_t)rp0 * NC + cidx];
            bv.y = xb[(size_t)rp1 * NC + cidx];
            CHUNK_MMA(abase + kk * 4, bv);
        }
    }

    if (c0 + n < NC) {
        // D layout: VGPR v -> row v (lanes 0-15) / row v+8 (lanes 16-31)
        float* ob = out + ((size_t)b * NL + l0 + kh * 8) * NC + c0 + n;
#pragma unroll
        for (int v = 0; v < 8; ++v) {
            ob[(size_t)(v)      * NC] = acc0[v];
            ob[(size_t)(v + 16) * NC] = acc1[v];
            ob[(size_t)(v + 32) * NC] = acc2[v];
            ob[(size_t)(v + 48) * NC] = acc3[v];
        }
    }
}

extern "C" void kernel_launch(void* const* d_in, const int* in_sizes, int n_in,
                              void* d_out, int out_size, void* d_ws, size_t ws_size,
                              hipStream_t stream) {
    (void)in_sizes; (void)n_in; (void)out_size; (void)ws_size;
    const float* x = (const float*)d_in[0];
    float* out = (float*)d_out;
    float* wc  = (float*)d_ws;                 // 769 floats of scratch

    build_wc<<<dim3(1), dim3(512), 0, stream>>>(wc);
    dog_wmma<<<dim3(NTILES / WAVES), dim3(256), 0, stream>>>(x, wc, out);
}